// EctChannelsTransform_39281770889251
// MI455X (gfx1250) — compile-verified
//
#include <hip/hip_runtime.h>
#include <hip/hip_bf16.h>

// ---------------------------------------------------------------------------
// ECT-with-channels transform for MI455X (gfx1250, wave32).
//   nh = x @ v            -> V_WMMA_F32_16X16X4_F32 (K=3 zero-padded to 4)
//   ecc = sigmoid(100*(lin - nh)) accumulated per (batch*4+channel, r, t)
//        sigmoid(z) = 0.5 + 0.5*tanh(z/2)  -> single V_TANH_F32 per element
//   normalize per (batch, channel) by max over (r, t)
// Accumulation loop is software-pipelined in groups of 4 points so the next
// group's ds_loads are in flight while the current group's tanh + ds_add_f32
// execute (DScnt retires in order -> partial waits instead of full drains).
// ---------------------------------------------------------------------------

typedef float v2f __attribute__((ext_vector_type(2)));
typedef float v8f __attribute__((ext_vector_type(8)));

#define NPTS_TOTAL 32768
#define DIMS       3
#define NT         64     // directions
#define RES        64
#define RC         8      // resolution rows handled per block
#define NBATCH     32
#define NCH        4
#define PCHUNK     64     // points per WMMA/accumulate chunk
#define HALF_SCALE 50.0f  // SCALE/2 for the tanh formulation
// SCALE * log2(e) for the exp2 fallback path
#define SCALE_LOG2E 144.26950408889634f

__device__ __forceinline__ float fast_sigmoid(float nh, float lin, float lin_hs) {
#if __has_builtin(__builtin_amdgcn_tanhf)
  // z/2 = HALF_SCALE*(lin - nh) = fma(-HALF_SCALE, nh, lin_hs)
  const float h = __builtin_amdgcn_tanhf(__builtin_fmaf(-HALF_SCALE, nh, lin_hs));
  return __builtin_fmaf(0.5f, h, 0.5f);
#else
  const float e = __builtin_amdgcn_exp2f(SCALE_LOG2E * (nh - lin));
  return __builtin_amdgcn_rcpf(1.0f + e);
#endif
}

__device__ __forceinline__ int lower_bound_i32(const int* __restrict__ a, int n, int v) {
  int lo = 0, hi = n;
  while (lo < hi) {
    int mid = (lo + hi) >> 1;
    if (a[mid] < v) lo = mid + 1; else hi = mid;
  }
  return lo;
}

// grid: (RES/RC, NBATCH), block: 512 threads (16 waves)
__global__ __launch_bounds__(512) void
ect_accum_kernel(const float* __restrict__ x, const float* __restrict__ v,
                 const int* __restrict__ index, const int* __restrict__ channels,
                 float* __restrict__ out, int n_points) {
  __shared__ float xs0[PCHUNK], xs1[PCHUNK], xs2[PCHUNK];
  __shared__ __align__(16) int chs[PCHUNK];
  __shared__ float nhs[PCHUNK * NT];        // 16 KB: projected chunk
  __shared__ float ect[NCH * RC * NT];      // 8 KB: per-block output slab
  __shared__ int   range[2];

  const int tid    = threadIdx.x;
  const int b      = blockIdx.y;
  const int rchunk = blockIdx.x;

  for (int i = tid; i < NCH * RC * NT; i += 512) ect[i] = 0.0f;

  if (tid == 0) {
    range[0] = lower_bound_i32(index, n_points, b);
    range[1] = lower_bound_i32(index, n_points, b + 1);
  }
  __syncthreads();
  const int start = range[0];
  const int end   = range[1];

  // ---- per-wave WMMA tile setup (hoisted: B tile from v is loop-invariant)
  const int  lane = tid & 31;
  const int  w    = tid >> 5;              // wave 0..15
  const int  p0   = (w >> 2) << 4;         // point-tile base (0,16,32,48)
  const int  t0   = (w & 3) << 4;          // theta-tile base (0,16,32,48)
  const bool hiHalf = (lane >= 16);        // lanes 16..31 carry K=2,3
  const int  tcol = t0 + (lane & 15);
  v2f btile;
  btile.x = hiHalf ? v[2 * NT + tcol] : v[0 * NT + tcol];  // K=0 / K=2
  btile.y = hiHalf ? 0.0f             : v[1 * NT + tcol];  // K=1 / K=3 (pad)

  // ---- accumulation-phase thread mapping: one (r,t) bin owner per thread
  const int t    = tid & 63;
  const int rloc = tid >> 6;               // 0..7
  const float lin = -1.0f + 2.0f * (float)(rchunk * RC + rloc) / (float)(RES - 1);
  const float lin_hs = HALF_SCALE * lin;   // precomputed 50*lin for the FMA
  float* const ect_rt = &ect[rloc * NT + t];   // + c*RC*NT selects channel

  for (int base = start; base < end; base += PCHUNK) {
    const int npts = min(PCHUNK, end - base);

    if (tid < PCHUNK) {
      float a0 = 0.f, a1 = 0.f, a2 = 0.f; int cc = 0;
      if (tid < npts) {
        const int p = base + tid;
        a0 = x[3 * p + 0]; a1 = x[3 * p + 1]; a2 = x[3 * p + 2];
        cc = channels[p];
      }
      xs0[tid] = a0; xs1[tid] = a1; xs2[tid] = a2; chs[tid] = cc;
    }
    __syncthreads();

    // ---- nh tile: D = A(16x4) x B(4x16), f32, one wave per 16x16 tile
    {
      const int row = p0 + (lane & 15);
      v2f atile;
      atile.x = hiHalf ? xs2[row] : xs0[row];   // K=0 / K=2
      atile.y = hiHalf ? 0.0f     : xs1[row];   // K=1 / K=3 (pad)
      v8f acc = {0.f, 0.f, 0.f, 0.f, 0.f, 0.f, 0.f, 0.f};
      acc = __builtin_amdgcn_wmma_f32_16x16x4_f32(
          /*neg_a=*/false, atile, /*neg_b=*/false, btile,
          /*c_mod=*/(short)0, acc, /*reuse_a=*/false, /*reuse_b=*/false);
#pragma unroll
      for (int j = 0; j < 8; ++j) {
        const int m = p0 + j + (hiHalf ? 8 : 0);
        nhs[m * NT + t0 + (lane & 15)] = acc[j];
      }
    }
    __syncthreads();

    // ---- sigmoid soft-binning; serialized point loop => each LDS address is
    //      touched by exactly one thread: contention-free ds_add_f32.
    //      Pipelined: group g+1's loads are issued before group g's atomics.
    {
      const float* const nhp = &nhs[t];
      const int ngroups = npts >> 2;
      float n0 = 0.f, n1 = 0.f, n2 = 0.f, n3 = 0.f;
      int4 c4 = make_int4(0, 0, 0, 0);
      if (ngroups > 0) {
        n0 = nhp[0 * NT]; n1 = nhp[1 * NT]; n2 = nhp[2 * NT]; n3 = nhp[3 * NT];
        c4 = *(const int4*)&chs[0];
      }
      for (int g = 0; g < ngroups; ++g) {
        float m0 = 0.f, m1 = 0.f, m2 = 0.f, m3 = 0.f;
        int4 cn = c4;
        if (g + 1 < ngroups) {                       // prefetch next group
          const float* const q = nhp + (g + 1) * 4 * NT;
          m0 = q[0 * NT]; m1 = q[1 * NT]; m2 = q[2 * NT]; m3 = q[3 * NT];
          cn = *(const int4*)&chs[(g + 1) * 4];
        }
        atomicAdd(ect_rt + c4.x * (RC * NT), fast_sigmoid(n0, lin, lin_hs));
        atomicAdd(ect_rt + c4.y * (RC * NT), fast_sigmoid(n1, lin, lin_hs));
        atomicAdd(ect_rt + c4.z * (RC * NT), fast_sigmoid(n2, lin, lin_hs));
        atomicAdd(ect_rt + c4.w * (RC * NT), fast_sigmoid(n3, lin, lin_hs));
        n0 = m0; n1 = m1; n2 = m2; n3 = m3; c4 = cn;
      }
      for (int p = ngroups << 2; p < npts; ++p) {    // tail (< 4 points)
        const float nh = nhp[p * NT];
        atomicAdd(ect_rt + chs[p] * (RC * NT), fast_sigmoid(nh, lin, lin_hs));
      }
    }
    __syncthreads();
  }

  // ---- writeback: block owns disjoint (b, r-chunk) slab of out
  for (int i = tid; i < NCH * RC * NT; i += 512) {
    const int cc = i >> 9;             // / (RC*NT)
    const int rl = (i >> 6) & (RC - 1);
    const int tt = i & 63;
    const int rg = rchunk * RC + rl;
    out[(((b * NCH + cc) * RES) + rg) * NT + tt] = ect[i];
  }
}

// grid: 128 (one per batch*channel), block: 256 threads
__global__ __launch_bounds__(256) void
ect_norm_kernel(float* __restrict__ out) {
  __shared__ float red[256];
  const int tid = threadIdx.x;
  const size_t base = (size_t)blockIdx.x * (RES * NT);

  float m = 0.0f;  // sums are non-negative
  for (int i = tid; i < RES * NT; i += 256) m = fmaxf(m, out[base + i]);
  red[tid] = m;
  __syncthreads();
  for (int s = 128; s > 0; s >>= 1) {
    if (tid < s) red[tid] = fmaxf(red[tid], red[tid + s]);
    __syncthreads();
  }
  const float amax = red[0];
  const float inv  = (amax == 0.0f) ? 1.0f : (1.0f / amax);
  for (int i = tid; i < RES * NT; i += 256) out[base + i] *= inv;
}

extern "C" void kernel_launch(void* const* d_in, const int* in_sizes, int n_in,
                              void* d_out, int out_size, void* d_ws, size_t ws_size,
                              hipStream_t stream) {
  (void)n_in; (void)out_size; (void)d_ws; (void)ws_size;
  const float* x        = (const float*)d_in[0];
  const float* v        = (const float*)d_in[1];
  const int*   index    = (const int*)d_in[2];
  const int*   channels = (const int*)d_in[3];
  float*       out      = (float*)d_out;
  const int n_points = in_sizes[0] / DIMS;

  dim3 grid1(RES / RC, NBATCH);
  ect_accum_kernel<<<grid1, 512, 0, stream>>>(x, v, index, channels, out, n_points);
  ect_norm_kernel<<<NBATCH * NCH, 256, 0, stream>>>(out);
}